// GraphConvpoolMPNN_block_v6_6889127543477
// MI455X (gfx1250) — compile-verified
//
#include <hip/hip_runtime.h>

// ---------------------------------------------------------------------------
// GraphConvpool MPNN for MI455X (gfx1250, wave32).
// All GEMMs use V_WMMA_F32_16X16X4_F32 (exact fp32). Softmax over the 512-wide
// adjacency rows is done online (flash-style) so A is never materialized.
// k_main uses double-buffered LDS staging (load-early / store-late) so the
// global loads for tile j+1 are in flight behind the 32-WMMA block for tile j.
// Wm / Wt weight matrices are staged to LDS once per block.
// Workspace use: ~62.5 MB (H + pre-BN out + small stats regions).
// ---------------------------------------------------------------------------

typedef float v2f __attribute__((ext_vector_type(2)));
typedef float v8f __attribute__((ext_vector_type(8)));

#define TWS    8
#define WSTRIDE 4
#define SENS   64
#define FDIM   64
#define TDIM   128
#define BDIM   8
#define NW     31            // (128-8)/4+1
#define NGR    (BDIM*NW)     // 248 graphs
#define NN     512           // TWS*SENS nodes per graph
#define NROWS  126976.0f     // NGR*NN  (== BDIM*SENS*sum(cnt_t))

__constant__ float c_DPOW[8] = {1.0f, 0.7f, 0.49f, 0.343f, 0.2401f,
                                0.16807f, 0.117649f, 0.0823543f};

// D = A(16x4 f32) * B(4x16 f32) + C(16x16 f32)   -- VOP3P V_WMMA_F32_16X16X4_F32
__device__ __forceinline__ v8f wmma4(v2f a, v2f b, v8f c) {
  return __builtin_amdgcn_wmma_f32_16x16x4_f32(false, a, false, b,
                                               (short)0, c, false, false);
}

// reductions across a 16-lane half of a wave32 (xor masks 1,2,4,8 stay in-half)
__device__ __forceinline__ float rmax16(float v) {
#pragma unroll
  for (int m = 1; m < 16; m <<= 1) v = fmaxf(v, __shfl_xor(v, m, 32));
  return v;
}
__device__ __forceinline__ float rsum16(float v) {
#pragma unroll
  for (int m = 1; m < 16; m <<= 1) v += __shfl_xor(v, m, 32);
  return v;
}

// ---------------------------------------------------------------------------
// BN0 stats: per (t, channel) partial sums over (B,S); weighted by window
// multiplicity in the finalize kernel. Deterministic (no atomics).
// ---------------------------------------------------------------------------
__global__ __launch_bounds__(256) void k_stats0(const float* __restrict__ x,
                                                float* __restrict__ psum,
                                                float* __restrict__ psq) {
  const int t = blockIdx.x;
  const int f = threadIdx.x & 63, part = threadIdx.x >> 6;
  float s = 0.f, q = 0.f;
  for (int r = part; r < BDIM * SENS; r += 4) {
    const int b = r >> 6, ss = r & 63;
    const float v = x[(((size_t)b * TDIM + t) * SENS + ss) * FDIM + f];
    s += v; q += v * v;
  }
  __shared__ float rs[256], rq[256];
  rs[threadIdx.x] = s; rq[threadIdx.x] = q;
  __syncthreads();
  if (part == 0) {
    for (int p = 1; p < 4; ++p) { s += rs[p * 64 + f]; q += rq[p * 64 + f]; }
    psum[t * 64 + f] = s; psq[t * 64 + f] = q;
  }
}

__global__ void k_fin0(const float* __restrict__ psum, const float* __restrict__ psq,
                       const float* __restrict__ g, const float* __restrict__ bb,
                       float* __restrict__ aC, float* __restrict__ cC) {
  const int f = threadIdx.x;  // 64 threads
  float s = 0.f, q = 0.f;
  for (int t = 0; t < TDIM; ++t) {
    const int lo = (t >= 7) ? ((t - 4) >> 2) : 0;     // ceil((t-7)/4) clamped
    const int hv = t >> 2;
    const int hi = hv < 30 ? hv : 30;
    const float c = (float)(hi - lo + 1);             // window multiplicity
    s += c * psum[t * 64 + f];
    q += c * psq[t * 64 + f];
  }
  const float inv = 1.0f / NROWS;
  const float mean = s * inv;
  const float var = q * inv - mean * mean;
  const float a = rsqrtf(var + 1e-5f) * g[f];
  aC[f] = a;
  cC[f] = bb[f] - mean * a;                           // Xb = x*a + c
}

// ---------------------------------------------------------------------------
// H = Xn @ Wm + bm  for all graphs.  1 wave = one 16x64 tile.
// Wm (16 KB) staged to LDS once per block; B-frags then come from LDS banks.
// ---------------------------------------------------------------------------
__global__ __launch_bounds__(128) void k_H(const float* __restrict__ x,
                                           const float* __restrict__ Wm,
                                           const float* __restrict__ bm,
                                           float* __restrict__ Hout) {
  __shared__ __align__(16) float Wms[FDIM * FDIM];
  const int tid = threadIdx.x;
#pragma unroll
  for (int i = 0; i < 8; ++i)
    *(float4*)&Wms[tid * 32 + i * 4] = *(const float4*)(Wm + tid * 32 + i * 4);

  const int lane = tid & 31, wave = tid >> 5;
  const int idx = blockIdx.x * 4 + wave;              // tile id
  const int bn = idx >> 5, it = idx & 31, ibase = it * 16;
  const int b = bn / NW, w = bn % NW;
  const int half = lane >> 4, l15 = lane & 15;

  const int n = ibase + l15;                          // node row (unfold)
  const int t = w * WSTRIDE + (n >> 6);
  const float* xrow = x + (((size_t)(b * TDIM + t) * SENS + (n & 63)) * FDIM);

  v2f A[16];                                          // A-frags: 16x64, K-steps of 4
#pragma unroll
  for (int kk = 0; kk < 16; ++kk)
    A[kk] = *(const v2f*)(xrow + kk * 4 + half * 2);

  __syncthreads();

  float* hrow = Hout + ((size_t)bn * NN + ibase) * FDIM;
#pragma unroll
  for (int nt = 0; nt < 4; ++nt) {
    const int ncol = nt * 16 + l15;
    const float bmv = bm[ncol];
    v8f acc;
#pragma unroll
    for (int v = 0; v < 8; ++v) acc[v] = bmv;
#pragma unroll
    for (int kk = 0; kk < 16; ++kk) {
      const int kb = kk * 4 + half * 2;
      v2f bf;
      bf.x = Wms[kb * FDIM + ncol];
      bf.y = Wms[(kb + 1) * FDIM + ncol];
      acc = wmma4(A[kk], bf, acc);
    }
#pragma unroll
    for (int v = 0; v < 8; ++v)
      hrow[(v + 8 * half) * FDIM + ncol] = acc[v];
  }
}

// ---------------------------------------------------------------------------
// Fused: scores=H@H^T -> leaky -> online softmax -> *decay-mask -> @Xb
//        -> (+Xb_i for the +eye term) -> @Wt + bt  -> pre-BN out.
// grid (248 graphs, 4 row-tile groups), 256 threads = 8 waves, 1 tile/wave.
// Double-buffered staging: tile j+1's global loads issue before tile j's
// WMMA block; LDS stores + one barrier per iteration close the pipeline.
// ---------------------------------------------------------------------------
__global__ __launch_bounds__(256) void k_main(const float* __restrict__ x,
                                              const float* __restrict__ Wt,
                                              const float* __restrict__ bt,
                                              const float* __restrict__ Hg,
                                              const float* __restrict__ aC,
                                              const float* __restrict__ cC,
                                              float* __restrict__ preout) {
  __shared__ __align__(16) float Hs[2][16 * 64];      // staged H_j tiles
  __shared__ __align__(16) float Xs[2][16 * 64];      // staged Xb_j tiles
  __shared__ __align__(16) float Ps[8][16 * 16];      // per-wave P relayout
  __shared__ __align__(16) float Os[8][16 * 64];      // per-wave O relayout
  __shared__ __align__(16) float Wts[FDIM * FDIM];    // staged Wt (16 KB)

  const int tid = threadIdx.x, lane = tid & 31, wave = tid >> 5;
  const int half = lane >> 4, l15 = lane & 15;
  const int bn = blockIdx.x, b = bn / NW, w = bn % NW;
  const float* Hb = Hg + (size_t)bn * NN * FDIM;

  // cooperative staging: each thread owns 4 consecutive floats of a 16x64 tile
  const int se = tid * 4;
  const int srow = se >> 6, scol = se & 63;
  const float4 a4 = *(const float4*)(aC + scol);
  const float4 c4 = *(const float4*)(cC + scol);

  const int it = blockIdx.y * 8 + wave, ibase = it * 16;

  // stage Wt once (256 threads x 16 floats)
#pragma unroll
  for (int i = 0; i < 4; ++i)
    *(float4*)&Wts[tid * 16 + i * 4] = *(const float4*)(Wt + tid * 16 + i * 4);

  // A-frags of this wave's H row tile (held in VGPRs for the whole j loop)
  v2f Afr[16];
  {
    const float* hirow = Hb + (size_t)(ibase + l15) * FDIM;
#pragma unroll
    for (int kk = 0; kk < 16; ++kk)
      Afr[kk] = *(const v2f*)(hirow + kk * 4 + half * 2);
  }

  v8f Oa[4], mrun, lrun;
#pragma unroll
  for (int v = 0; v < 8; ++v) { mrun[v] = -1e30f; lrun[v] = 0.0f; }
#pragma unroll
  for (int nt = 0; nt < 4; ++nt)
#pragma unroll
    for (int v = 0; v < 8; ++v) Oa[nt][v] = 0.0f;

  // ---- prologue: stage tile j=0 into buffer 0 ----
  {
    const float4 h0 = *(const float4*)(Hb + (size_t)srow * FDIM + scol);
    const int tt0 = w * WSTRIDE + (srow >> 6);        // jbase=0 -> nrow=srow
    const float4 x0 = *(const float4*)(
        x + (((size_t)(b * TDIM + tt0) * SENS + (srow & 63)) * FDIM) + scol);
    *(float4*)&Hs[0][se] = h0;
    float4 o;
    o.x = x0.x * a4.x + c4.x; o.y = x0.y * a4.y + c4.y;
    o.z = x0.z * a4.z + c4.z; o.w = x0.w * a4.w + c4.w;
    *(float4*)&Xs[0][se] = o;
  }
  __syncthreads();

  for (int j = 0; j < 32; ++j) {
    const int jbase = j * 16;
    const int cur = j & 1;

    // ---- issue next tile's global loads before this tile's compute ----
    float4 hn, xn;
    if (j < 31) {
      const int nrow = jbase + 16 + srow;
      hn = *(const float4*)(Hb + (size_t)nrow * FDIM + scol);
      const int tt = w * WSTRIDE + (nrow >> 6);
      xn = *(const float4*)(
          x + (((size_t)(b * TDIM + tt) * SENS + (nrow & 63)) * FDIM) + scol);
    }

    // ---- scores: 16x16 tile of H_i @ H_j^T, K=64 ----
    v8f s;
#pragma unroll
    for (int v = 0; v < 8; ++v) s[v] = 0.0f;
#pragma unroll
    for (int kk = 0; kk < 16; ++kk) {
      const int kb = kk * 4 + half * 2;
      v2f bf = *(const v2f*)&Hs[cur][l15 * FDIM + kb]; // B[k][n]=H[jbase+n][k]
      s = wmma4(Afr[kk], bf, s);
    }

    int d = (ibase >> 6) - (jbase >> 6); d = d < 0 ? -d : d;
    const float maskc = c_DPOW[d];                    // mask const per tile pair

    // ---- leaky + diag kill + online softmax update ----
    v8f e, rm;
#pragma unroll
    for (int v = 0; v < 8; ++v) {
      float sv = s[v];
      sv = sv >= 0.0f ? sv : 0.01f * sv;              // leaky_relu
      if (ibase == jbase && l15 == v + 8 * half) sv = -1e30f;  // -1e8 diag -> 0
      s[v] = sv;
      rm[v] = rmax16(sv);
    }
#pragma unroll
    for (int v = 0; v < 8; ++v) {
      const float mn = fmaxf(mrun[v], rm[v]);
      const float sc = __expf(mrun[v] - mn);
      const float ev = __expf(s[v] - mn);
      e[v] = ev;
      lrun[v] = lrun[v] * sc + rsum16(ev);            // denom excludes mask
      mrun[v] = mn;
#pragma unroll
      for (int nt = 0; nt < 4; ++nt) Oa[nt][v] *= sc; // rescale running O
    }

    // ---- relayout P (C-layout -> A-layout) via per-wave LDS ----
    float* Pw = Ps[wave];
#pragma unroll
    for (int v = 0; v < 8; ++v)
      Pw[(v + 8 * half) * 16 + l15] = e[v] * maskc;
    v2f Pf[4];
#pragma unroll
    for (int kk = 0; kk < 4; ++kk)
      Pf[kk] = *(const v2f*)&Pw[l15 * 16 + kk * 4 + half * 2];

    // ---- O += P @ Xb_j  (16x16 * 16x64) ----
#pragma unroll
    for (int nt = 0; nt < 4; ++nt) {
#pragma unroll
      for (int kk = 0; kk < 4; ++kk) {
        const int kb = kk * 4 + half * 2;
        v2f bf;
        bf.x = Xs[cur][kb * FDIM + nt * 16 + l15];
        bf.y = Xs[cur][(kb + 1) * FDIM + nt * 16 + l15];
        Oa[nt] = wmma4(Pf[kk], bf, Oa[nt]);
      }
    }

    // ---- store next tile into the other buffer, then barrier ----
    if (j < 31) {
      *(float4*)&Hs[cur ^ 1][se] = hn;
      float4 o;
      o.x = xn.x * a4.x + c4.x; o.y = xn.y * a4.y + c4.y;
      o.z = xn.z * a4.z + c4.z; o.w = xn.w * a4.w + c4.w;
      *(float4*)&Xs[cur ^ 1][se] = o;
    }
    __syncthreads();
  }

  // ---- finalize: (O/l + Xb_i) @ Wt + bt -> preout tile ----
  float* Ow = Os[wave];
#pragma unroll
  for (int v = 0; v < 8; ++v) {
    const float inv = 1.0f / lrun[v];
#pragma unroll
    for (int nt = 0; nt < 4; ++nt)
      Ow[(v + 8 * half) * FDIM + nt * 16 + l15] = Oa[nt][v] * inv;
  }
  const int nrow = ibase + l15;
  const int tt = w * WSTRIDE + (nrow >> 6);
  const float* xi = x + (((size_t)(b * TDIM + tt) * SENS + (nrow & 63)) * FDIM);
  v2f A2[16];
#pragma unroll
  for (int kk = 0; kk < 16; ++kk) {
    const int kb = kk * 4 + half * 2;
    v2f ov = *(const v2f*)&Ow[l15 * FDIM + kb];
    v2f xv = *(const v2f*)(xi + kb);
    A2[kk].x = ov.x + xv.x * aC[kb] + cC[kb];         // + eye*Xb term
    A2[kk].y = ov.y + xv.y * aC[kb + 1] + cC[kb + 1];
  }
  float* prow = preout + ((size_t)bn * NN + ibase) * FDIM;
#pragma unroll
  for (int nt = 0; nt < 4; ++nt) {
    const int ncol = nt * 16 + l15;
    const float btv = bt[ncol];
    v8f acc;
#pragma unroll
    for (int v = 0; v < 8; ++v) acc[v] = btv;
#pragma unroll
    for (int kk = 0; kk < 16; ++kk) {
      const int kb = kk * 4 + half * 2;
      v2f bf;
      bf.x = Wts[kb * FDIM + ncol];
      bf.y = Wts[(kb + 1) * FDIM + ncol];
      acc = wmma4(A2[kk], bf, acc);
    }
#pragma unroll
    for (int v = 0; v < 8; ++v)
      prow[(v + 8 * half) * FDIM + ncol] = acc[v];
  }
}

// ---------------------------------------------------------------------------
// BN1 stats over preout, then BN1 + leaky + mean-pool over TWS.
// ---------------------------------------------------------------------------
__global__ __launch_bounds__(256) void k_stats1(const float* __restrict__ pre,
                                                float* __restrict__ psum,
                                                float* __restrict__ psq) {
  const int bn = blockIdx.x;
  const int f = threadIdx.x & 63, part = threadIdx.x >> 6;
  float s = 0.f, q = 0.f;
  for (int r = part; r < NN; r += 4) {
    const float v = pre[((size_t)bn * NN + r) * FDIM + f];
    s += v; q += v * v;
  }
  __shared__ float rs[256], rq[256];
  rs[threadIdx.x] = s; rq[threadIdx.x] = q;
  __syncthreads();
  if (part == 0) {
    for (int p = 1; p < 4; ++p) { s += rs[p * 64 + f]; q += rq[p * 64 + f]; }
    psum[bn * 64 + f] = s; psq[bn * 64 + f] = q;
  }
}

__global__ void k_fin1(const float* __restrict__ psum, const float* __restrict__ psq,
                       const float* __restrict__ g, const float* __restrict__ bb,
                       float* __restrict__ a1, float* __restrict__ c1) {
  const int f = threadIdx.x;  // 64 threads
  float s = 0.f, q = 0.f;
  for (int p = 0; p < NGR; ++p) { s += psum[p * 64 + f]; q += psq[p * 64 + f]; }
  const float inv = 1.0f / NROWS;
  const float mean = s * inv;
  const float var = q * inv - mean * mean;
  const float a = rsqrtf(var + 1e-5f) * g[f];
  a1[f] = a;
  c1[f] = bb[f] - mean * a;
}

__global__ __launch_bounds__(256) void k_out(const float* __restrict__ pre,
                                             const float* __restrict__ a1,
                                             const float* __restrict__ c1,
                                             float* __restrict__ out) {
  const int idx = blockIdx.x * 256 + threadIdx.x;
  if (idx >= BDIM * NW * SENS * FDIM) return;
  const int c = idx & 63;
  const int s = (idx >> 6) & 63;
  const int bw = idx >> 12;
  const int w = bw % NW, b = bw / NW;
  const int bn = b * NW + w;
  const float a = a1[c], cc = c1[c];
  float acc = 0.f;
#pragma unroll
  for (int tt = 0; tt < TWS; ++tt) {
    const float y = pre[((size_t)bn * NN + tt * SENS + s) * FDIM + c] * a + cc;
    acc += (y >= 0.f ? y : 0.01f * y);
  }
  out[idx] = acc * 0.125f;
}

// ---------------------------------------------------------------------------
extern "C" void kernel_launch(void* const* d_in, const int* in_sizes, int n_in,
                              void* d_out, int out_size, void* d_ws, size_t ws_size,
                              hipStream_t stream) {
  const float* x  = (const float*)d_in[0];
  const float* Wm = (const float*)d_in[1];
  const float* bm = (const float*)d_in[2];
  const float* g0 = (const float*)d_in[3];
  const float* b0 = (const float*)d_in[4];
  const float* Wt = (const float*)d_in[5];
  const float* bt = (const float*)d_in[6];
  const float* g1 = (const float*)d_in[7];
  const float* b1 = (const float*)d_in[8];

  float* ws  = (float*)d_ws;
  float* aC  = ws + 0;                       // [64]
  float* cC  = ws + 64;                      // [64]
  float* a1  = ws + 128;                     // [64]
  float* c1  = ws + 192;                     // [64]
  float* ps0 = ws + 256;                     // [128*64]
  float* pq0 = ps0 + 128 * 64;               // [128*64]
  float* ps1 = pq0 + 128 * 64;               // [248*64]
  float* pq1 = ps1 + NGR * 64;               // [248*64]
  float* Hg  = pq1 + NGR * 64;               // [248*512*64]  (32.5 MB)
  float* pre = Hg + (size_t)NGR * NN * FDIM; // [248*512*64]  (32.5 MB)

  k_stats0<<<TDIM, 256, 0, stream>>>(x, ps0, pq0);
  k_fin0<<<1, 64, 0, stream>>>(ps0, pq0, g0, b0, aC, cC);
  k_H<<<(NGR * 32) / 4, 128, 0, stream>>>(x, Wm, bm, Hg);
  k_main<<<dim3(NGR, 4), 256, 0, stream>>>(x, Wt, bt, Hg, aC, cC, pre);
  k_stats1<<<NGR, 256, 0, stream>>>(pre, ps1, pq1);
  k_fin1<<<1, 64, 0, stream>>>(ps1, pq1, g1, b1, a1, c1);
  const int tot = BDIM * NW * SENS * FDIM;
  k_out<<<(tot + 255) / 256, 256, 0, stream>>>(pre, a1, c1, (float*)d_out);
}